// Net_58007828300427
// MI455X (gfx1250) — compile-verified
//
#include <hip/hip_runtime.h>

typedef __attribute__((ext_vector_type(2))) float v2f;
typedef __attribute__((ext_vector_type(8))) float v8f;

#define BM 128
#define BN 128
#define TK 16
#define HP 256           // padded feature stride (H=200 zero-padded)
#define LDSTR 20         // padded LDS column/row stride (bank-conflict-free b64 frags)
#define TILEF (128 * LDSTR)

// ---------------- CDNA5 async Global->LDS (Tensor-ish DMA path) ----------------
__device__ __forceinline__ void async_ld_b128(unsigned lds_addr, const float* g) {
  asm volatile("global_load_async_to_lds_b128 %0, %1, off" :: "v"(lds_addr), "v"(g) : "memory");
}
__device__ __forceinline__ void async_ld_b32(unsigned lds_addr, const float* g) {
  asm volatile("global_load_async_to_lds_b32 %0, %1, off" :: "v"(lds_addr), "v"(g) : "memory");
}

// =====================================================================
// C = A(MxK) @ B(KxN), exact fp32 via V_WMMA_F32_16X16X4_F32.
// Block = 256 threads = 8 waves arranged 4(M) x 2(N); block tile 128x128;
// wave tile 32x64 (8 v8f accumulators). Double-buffered LDS tiles filled
// with global_load_async_to_lds_*, synchronized via s_wait_asynccnt.
// A tile: row-major, row stride 20.  B tile: k-pair interleaved,
// sB[n*20 + (k>>1)*2 + (k&1)] so a B fragment is one ds_load_b64.
// All M%128==0, N%128==0, K%16==0 by construction.
// =====================================================================
__global__ __launch_bounds__(256) void gemm_wmma_f32(
    const float* __restrict__ A, const float* __restrict__ B,
    float* __restrict__ C, int K, int lda, int ldb, int ldc)
{
  __shared__ __align__(16) float sA[2][TILEF];
  __shared__ __align__(16) float sB[2][TILEF];

  const int tid  = threadIdx.x;
  const int wave = tid >> 5;
  const int lane = tid & 31;
  const int half = lane >> 4;     // 0: lanes 0-15, 1: lanes 16-31
  const int l16  = lane & 15;
  const int wm   = wave >> 1;     // 0..3  (M strip of 32 rows)
  const int wn   = wave & 1;      // 0..1  (N strip of 64 cols)
  const int bm   = blockIdx.y * BM;
  const int bn   = blockIdx.x * BN;

  const unsigned sA0 = (unsigned)(size_t)(void*)&sA[0][0];
  const unsigned sB0 = (unsigned)(size_t)(void*)&sB[0][0];

  v8f acc00 = {}, acc01 = {}, acc02 = {}, acc03 = {};
  v8f acc10 = {}, acc11 = {}, acc12 = {}, acc13 = {};

  // ---- async tile issue: 2x b128 (A) + 8x b32 (B, interleaving layout) ----
  auto issue_tile = [&](int buf, int k0) {
#pragma unroll
    for (int j = 0; j < 2; ++j) {           // A: 128x16, 512 b128 chunks
      int chunk = j * 256 + tid;
      int row = chunk >> 2, seg = chunk & 3;
      async_ld_b128(sA0 + (unsigned)((buf * TILEF + row * LDSTR + seg * 4) * 4),
                    A + (size_t)(bm + row) * lda + (k0 + seg * 4));
    }
#pragma unroll
    for (int j = 0; j < 8; ++j) {           // B: 16x128 -> pair-interleaved
      int e = j * 256 + tid;
      int r = e >> 7, c = e & 127;          // consecutive lanes -> consecutive cols
      async_ld_b32(sB0 + (unsigned)((buf * TILEF + c * LDSTR + (r >> 1) * 2 + (r & 1)) * 4),
                   B + (size_t)(k0 + r) * ldb + (bn + c));
    }
  };

  const int nt = K / TK;
  issue_tile(0, 0);

  for (int i = 0; i < nt; ++i) {
    const int cur = i & 1;
    if (i + 1 < nt) {
      issue_tile(cur ^ 1, (i + 1) * TK);
      asm volatile("s_wait_asynccnt 10" ::: "memory");  // tile i landed; tile i+1 in flight
    } else {
      asm volatile("s_wait_asynccnt 0" ::: "memory");
    }
    __syncthreads();

    const float* __restrict__ tA = &sA[cur][0];
    const float* __restrict__ tB = &sB[cur][0];
#pragma unroll
    for (int ks = 0; ks < 4; ++ks) {
      // A frags 16x4: lanes 0-15 -> K=ks*4+{0,1}; lanes 16-31 -> K=ks*4+{2,3}
      v2f a0 = *(const v2f*)&tA[(wm * 32 +      l16) * LDSTR + ks * 4 + 2 * half];
      v2f a1 = *(const v2f*)&tA[(wm * 32 + 16 + l16) * LDSTR + ks * 4 + 2 * half];
      // B frags 4x16: one b64 each thanks to pair-interleaved layout
      const int pp = (ks * 2 + half) * 2;
      v2f b0 = *(const v2f*)&tB[(wn * 64 +      l16) * LDSTR + pp];
      v2f b1 = *(const v2f*)&tB[(wn * 64 + 16 + l16) * LDSTR + pp];
      v2f b2 = *(const v2f*)&tB[(wn * 64 + 32 + l16) * LDSTR + pp];
      v2f b3 = *(const v2f*)&tB[(wn * 64 + 48 + l16) * LDSTR + pp];
      acc00 = __builtin_amdgcn_wmma_f32_16x16x4_f32(false, a0, false, b0, (short)0, acc00, false, false);
      acc01 = __builtin_amdgcn_wmma_f32_16x16x4_f32(false, a0, false, b1, (short)0, acc01, false, false);
      acc02 = __builtin_amdgcn_wmma_f32_16x16x4_f32(false, a0, false, b2, (short)0, acc02, false, false);
      acc03 = __builtin_amdgcn_wmma_f32_16x16x4_f32(false, a0, false, b3, (short)0, acc03, false, false);
      acc10 = __builtin_amdgcn_wmma_f32_16x16x4_f32(false, a1, false, b0, (short)0, acc10, false, false);
      acc11 = __builtin_amdgcn_wmma_f32_16x16x4_f32(false, a1, false, b1, (short)0, acc11, false, false);
      acc12 = __builtin_amdgcn_wmma_f32_16x16x4_f32(false, a1, false, b2, (short)0, acc12, false, false);
      acc13 = __builtin_amdgcn_wmma_f32_16x16x4_f32(false, a1, false, b3, (short)0, acc13, false, false);
    }
    __syncthreads();
  }

  // C/D layout: VGPR r, lanes 0-15 -> M=r, lanes 16-31 -> M=r+8; N=l16
  const int col = bn + wn * 64 + l16;
  {
    float* cp = C + (size_t)(bm + wm * 32 + 8 * half) * ldc + col;
#pragma unroll
    for (int r = 0; r < 8; ++r) {
      cp[(size_t)r * ldc +  0] = acc00[r];
      cp[(size_t)r * ldc + 16] = acc01[r];
      cp[(size_t)r * ldc + 32] = acc02[r];
      cp[(size_t)r * ldc + 48] = acc03[r];
    }
  }
  {
    float* cp = C + (size_t)(bm + wm * 32 + 16 + 8 * half) * ldc + col;
#pragma unroll
    for (int r = 0; r < 8; ++r) {
      cp[(size_t)r * ldc +  0] = acc10[r];
      cp[(size_t)r * ldc + 16] = acc11[r];
      cp[(size_t)r * ldc + 32] = acc12[r];
      cp[(size_t)r * ldc + 48] = acc13[r];
    }
  }
}

// ============================ elementwise ============================

__global__ void k_pad_mat(float* __restrict__ dst, const float* __restrict__ src,
                          int rsrc, int csrc, int total) {
  int idx = blockIdx.x * 256 + threadIdx.x;
  if (idx >= total) return;
  int r = idx >> 8, c = idx & 255;
  dst[idx] = (r < rsrc && c < csrc) ? src[r * csrc + c] : 0.0f;
}

__global__ void k_pad_x(float* __restrict__ dst, const float* __restrict__ src, int n) {
  int idx = blockIdx.x * 256 + threadIdx.x;
  if (idx >= n * 16) return;
  int r = idx >> 4, c = idx & 15;
  dst[idx] = (c < 3) ? src[r * 3 + c] : 0.0f;
}

// dinv[row] = rsqrt(2 + sum_c A[row,c])   (A diag == 0 in all uses; Ah = A + 2I)
__global__ void k_rowsum_dinv(const float* __restrict__ A, float* __restrict__ dinv, int n) {
  __shared__ float red[256];
  const float* ar = A + (size_t)blockIdx.x * n;
  float s = 0.f;
  for (int c = threadIdx.x; c < n; c += 256) s += ar[c];
  red[threadIdx.x] = s;
  __syncthreads();
  for (int o = 128; o > 0; o >>= 1) {
    if (threadIdx.x < o) red[threadIdx.x] += red[threadIdx.x + o];
    __syncthreads();
  }
  if (threadIdx.x == 0) dinv[blockIdx.x] = rsqrtf(red[0] + 2.0f);
}

// An = dinv_r * (A + 2I) * dinv_c
__global__ void k_normalize(const float* __restrict__ A, float* __restrict__ An,
                            const float* __restrict__ dinv, int n) {
  size_t idx = (size_t)blockIdx.x * 256 + threadIdx.x;
  size_t r = idx / (size_t)n, c = idx - r * (size_t)n;
  float v = A[idx] + ((r == c) ? 2.0f : 0.0f);
  An[idx] = dinv[r] * v * dinv[c];
}

// augment(A) = offdiag(A@A + 2A); C already holds A@A
__global__ void k_augment_post(float* __restrict__ C, const float* __restrict__ A, int n) {
  size_t idx = (size_t)blockIdx.x * 256 + threadIdx.x;
  size_t r = idx / (size_t)n, c = idx - r * (size_t)n;
  C[idx] = (r == c) ? 0.0f : C[idx] + 2.0f * A[idx];
}

// Y += bias (c<hc), optional relu; zero padded columns
__global__ void k_gcn_post(float* __restrict__ Y, const float* __restrict__ b,
                           int hc, int relu, int total) {
  int idx = blockIdx.x * 256 + threadIdx.x;
  if (idx >= total) return;
  int c = idx & 255;
  float v = 0.0f;
  if (c < hc) {
    v = Y[idx] + b[c];
    if (relu) v = fmaxf(v, 0.0f);
  }
  Y[idx] = v;
}

__global__ void k_vecnorm_inv(const float* __restrict__ p, int len, float* __restrict__ out) {
  __shared__ float red[256];
  float s = 0.f;
  for (int c = threadIdx.x; c < len; c += 256) s += p[c] * p[c];
  red[threadIdx.x] = s;
  __syncthreads();
  for (int o = 128; o > 0; o >>= 1) {
    if (threadIdx.x < o) red[threadIdx.x] += red[threadIdx.x + o];
    __syncthreads();
  }
  if (threadIdx.x == 0) out[0] = rsqrtf(red[0]);
}

__global__ void k_score(const float* __restrict__ X, const float* __restrict__ p,
                        const float* __restrict__ inv, float* __restrict__ s, int n) {
  int i = blockIdx.x * 256 + threadIdx.x;
  if (i >= n) return;
  const float* xr = X + (size_t)i * HP;
  float d = 0.f;
  for (int c = 0; c < 200; ++c) d += xr[c] * p[c];
  s[i] = tanhf(d * inv[0]);
}

// exact jax.lax.top_k: stable descending via rank counting (unique ranks)
__global__ void k_topk(const float* __restrict__ s, int n, int k,
                       int* __restrict__ perm, float* __restrict__ val) {
  int i = blockIdx.x * 256 + threadIdx.x;
  if (i >= n) return;
  float si = s[i];
  int rank = 0;
  for (int j = 0; j < n; ++j) {
    float sj = s[j];
    rank += (sj > si) || (sj == si && j < i);
  }
  if (rank < k) { perm[rank] = i; val[rank] = si; }
}

__global__ void k_gather_A(const float* __restrict__ Aold, int nold,
                           const int* __restrict__ perm, float* __restrict__ Anew, int k) {
  size_t idx = (size_t)blockIdx.x * 256 + threadIdx.x;
  size_t r = idx / (size_t)k, c = idx - r * (size_t)k;
  Anew[idx] = Aold[(size_t)perm[r] * nold + perm[c]];
}

__global__ void k_gather_X(const float* __restrict__ Xold, const int* __restrict__ perm,
                           const float* __restrict__ val, float* __restrict__ Xnew, int k) {
  int idx = blockIdx.x * 256 + threadIdx.x;   // k * HP, exact grid
  int r = idx >> 8, c = idx & 255;
  Xnew[idx] = Xold[(size_t)perm[r] * HP + c] * val[r];
}

__global__ void k_scatter_add(float* __restrict__ dst, const float* __restrict__ xsrc,
                              const int* __restrict__ perm, int k) {
  int idx = blockIdx.x * 256 + threadIdx.x;   // k * HP, exact grid
  int r = idx >> 8, c = idx & 255;
  dst[(size_t)perm[r] * HP + c] += xsrc[idx];
}

__global__ void k_logsoftmax2(const float* __restrict__ Y, float* __restrict__ out, int n) {
  int i = blockIdx.x * 256 + threadIdx.x;
  if (i >= n) return;
  float y0 = Y[(size_t)i * HP + 0], y1 = Y[(size_t)i * HP + 1];
  float m = fmaxf(y0, y1);
  float l = m + logf(expf(y0 - m) + expf(y1 - m));
  out[2 * i]     = y0 - l;
  out[2 * i + 1] = y1 - l;
}

// ============================== host =================================

static inline void run_gemm(const float* A, const float* B, float* C,
                            int M, int N, int K, int lda, int ldb, int ldc,
                            hipStream_t s) {
  dim3 grid((unsigned)(N / BN), (unsigned)(M / BM));
  gemm_wmma_f32<<<grid, 256, 0, s>>>(A, B, C, K, lda, ldb, ldc);
}

extern "C" void kernel_launch(void* const* d_in, const int* in_sizes, int n_in,
                              void* d_out, int out_size, void* d_ws, size_t ws_size,
                              hipStream_t stream) {
  (void)in_sizes; (void)n_in; (void)out_size; (void)ws_size;
  const float* x   = (const float*)d_in[0];
  const float* adj = (const float*)d_in[1];
  const float* w0  = (const float*)d_in[2];
  const float* b0  = (const float*)d_in[3];
  const float* w1  = (const float*)d_in[4];
  const float* b1  = (const float*)d_in[5];
  const float* w2  = (const float*)d_in[6];
  const float* b2  = (const float*)d_in[7];
  const float* w3  = (const float*)d_in[8];
  const float* b3  = (const float*)d_in[9];
  const float* p1  = (const float*)d_in[10];
  const float* p2  = (const float*)d_in[11];
  const float* p3  = (const float*)d_in[12];
  const float* u0w = (const float*)d_in[13];
  const float* u0b = (const float*)d_in[14];
  const float* u1w = (const float*)d_in[15];
  const float* u1b = (const float*)d_in[16];
  const float* u2w = (const float*)d_in[17];
  const float* u2b = (const float*)d_in[18];
  float* out = (float*)d_out;

  const int N0 = 4096, K1 = 3072, K2 = 1536, K3 = 768;

  float* ws = (float*)d_ws;
  size_t off = 0;
  auto alloc = [&](size_t nf) { float* p = ws + off; off += (nf + 255) & ~(size_t)255; return p; };

  float* AN0 = alloc((size_t)N0 * N0);
  float* AUG = alloc((size_t)N0 * N0);
  float* A1  = alloc((size_t)K1 * K1);
  float* AN1 = alloc((size_t)K1 * K1);
  float* A2  = alloc((size_t)K2 * K2);
  float* AN2 = alloc((size_t)K2 * K2);
  float* A3  = alloc((size_t)K3 * K3);
  float* AN3 = alloc((size_t)K3 * K3);
  float* X0  = alloc((size_t)N0 * HP);
  float* X1  = alloc((size_t)K1 * HP);
  float* X2  = alloc((size_t)K2 * HP);
  float* X3  = alloc((size_t)K3 * HP);
  float* XW  = alloc((size_t)N0 * HP);
  float* UA  = alloc((size_t)N0 * HP);
  float* UB  = alloc((size_t)N0 * HP);
  float* XP  = alloc((size_t)N0 * 16);
  float* W0P = alloc(16 * HP);
  float* W1P = alloc(HP * HP);
  float* W2P = alloc(HP * HP);
  float* W3P = alloc(HP * HP);
  float* U0P = alloc(HP * HP);
  float* U1P = alloc(HP * HP);
  float* U2P = alloc(HP * HP);
  float* SC   = alloc(N0);
  float* VAL  = alloc(N0);
  int*   PERM1 = (int*)alloc(N0);
  int*   PERM2 = (int*)alloc(N0);
  int*   PERM3 = (int*)alloc(N0);
  float* DINV = alloc(N0);
  float* INV  = alloc(256);

  // ---- pad inputs / weights into WMMA-friendly shapes ----
  k_pad_x<<<(N0 * 16) / 256, 256, 0, stream>>>(XP, x, N0);
  k_pad_mat<<<(16 * HP) / 256, 256, 0, stream>>>(W0P, w0, 3, 200, 16 * HP);
  k_pad_mat<<<(HP * HP) / 256, 256, 0, stream>>>(W1P, w1, 200, 200, HP * HP);
  k_pad_mat<<<(HP * HP) / 256, 256, 0, stream>>>(W2P, w2, 200, 200, HP * HP);
  k_pad_mat<<<(HP * HP) / 256, 256, 0, stream>>>(W3P, w3, 200, 200, HP * HP);
  k_pad_mat<<<(HP * HP) / 256, 256, 0, stream>>>(U0P, u0w, 200, 200, HP * HP);
  k_pad_mat<<<(HP * HP) / 256, 256, 0, stream>>>(U1P, u1w, 200, 200, HP * HP);
  k_pad_mat<<<(HP * HP) / 256, 256, 0, stream>>>(U2P, u2w, 200, 2, HP * HP);

  // ---- level 0: An0 + gcn0 ----
  k_rowsum_dinv<<<N0, 256, 0, stream>>>(adj, DINV, N0);
  k_normalize<<<(unsigned)(((size_t)N0 * N0) / 256), 256, 0, stream>>>(adj, AN0, DINV, N0);
  run_gemm(XP, W0P, XW, N0, HP, 16, 16, HP, HP, stream);
  run_gemm(AN0, XW, X0, N0, HP, N0, N0, HP, HP, stream);
  k_gcn_post<<<(N0 * HP) / 256, 256, 0, stream>>>(X0, b0, 200, 1, N0 * HP);

  // ---- augment level 0: AUG = offdiag(adj@adj + 2adj) ----
  run_gemm(adj, adj, AUG, N0, N0, N0, N0, N0, N0, stream);
  k_augment_post<<<(unsigned)(((size_t)N0 * N0) / 256), 256, 0, stream>>>(AUG, adj, N0);

  // ---- pool 1 ----
  k_vecnorm_inv<<<1, 256, 0, stream>>>(p1, 200, INV);
  k_score<<<N0 / 256, 256, 0, stream>>>(X0, p1, INV, SC, N0);
  k_topk<<<N0 / 256, 256, 0, stream>>>(SC, N0, K1, PERM1, VAL);
  k_gather_A<<<(unsigned)(((size_t)K1 * K1) / 256), 256, 0, stream>>>(AUG, N0, PERM1, A1, K1);
  k_gather_X<<<(K1 * HP) / 256, 256, 0, stream>>>(X0, PERM1, VAL, X1, K1);

  // ---- gcn1 ----
  k_rowsum_dinv<<<K1, 256, 0, stream>>>(A1, DINV, K1);
  k_normalize<<<(unsigned)(((size_t)K1 * K1) / 256), 256, 0, stream>>>(A1, AN1, DINV, K1);
  run_gemm(X1, W1P, XW, K1, HP, HP, HP, HP, HP, stream);
  run_gemm(AN1, XW, X1, K1, HP, K1, K1, HP, HP, stream);
  k_gcn_post<<<(K1 * HP) / 256, 256, 0, stream>>>(X1, b1, 200, 1, K1 * HP);

  // ---- augment level 1 ----
  run_gemm(A1, A1, AUG, K1, K1, K1, K1, K1, K1, stream);
  k_augment_post<<<(unsigned)(((size_t)K1 * K1) / 256), 256, 0, stream>>>(AUG, A1, K1);

  // ---- pool 2 ----
  k_vecnorm_inv<<<1, 256, 0, stream>>>(p2, 200, INV);
  k_score<<<K1 / 256, 256, 0, stream>>>(X1, p2, INV, SC, K1);
  k_topk<<<K1 / 256, 256, 0, stream>>>(SC, K1, K2, PERM2, VAL);
  k_gather_A<<<(unsigned)(((size_t)K2 * K2) / 256), 256, 0, stream>>>(AUG, K1, PERM2, A2, K2);
  k_gather_X<<<(K2 * HP) / 256, 256, 0, stream>>>(X1, PERM2, VAL, X2, K2);

  // ---- gcn2 ----
  k_rowsum_dinv<<<K2, 256, 0, stream>>>(A2, DINV, K2);
  k_normalize<<<(unsigned)(((size_t)K2 * K2) / 256), 256, 0, stream>>>(A2, AN2, DINV, K2);
  run_gemm(X2, W2P, XW, K2, HP, HP, HP, HP, HP, stream);
  run_gemm(AN2, XW, X2, K2, HP, K2, K2, HP, HP, stream);
  k_gcn_post<<<(K2 * HP) / 256, 256, 0, stream>>>(X2, b2, 200, 1, K2 * HP);

  // ---- augment level 2 ----
  run_gemm(A2, A2, AUG, K2, K2, K2, K2, K2, K2, stream);
  k_augment_post<<<(unsigned)(((size_t)K2 * K2) / 256), 256, 0, stream>>>(AUG, A2, K2);

  // ---- pool 3 ----
  k_vecnorm_inv<<<1, 256, 0, stream>>>(p3, 200, INV);
  k_score<<<K2 / 256, 256, 0, stream>>>(X2, p3, INV, SC, K2);
  k_topk<<<K2 / 256, 256, 0, stream>>>(SC, K2, K3, PERM3, VAL);
  k_gather_A<<<(unsigned)(((size_t)K3 * K3) / 256), 256, 0, stream>>>(AUG, K2, PERM3, A3, K3);
  k_gather_X<<<(K3 * HP) / 256, 256, 0, stream>>>(X2, PERM3, VAL, X3, K3);

  // ---- gcn3 (bottom) ----
  k_rowsum_dinv<<<K3, 256, 0, stream>>>(A3, DINV, K3);
  k_normalize<<<(unsigned)(((size_t)K3 * K3) / 256), 256, 0, stream>>>(A3, AN3, DINV, K3);
  run_gemm(X3, W3P, XW, K3, HP, HP, HP, HP, HP, stream);
  run_gemm(AN3, XW, X3, K3, HP, K3, K3, HP, HP, stream);
  k_gcn_post<<<(K3 * HP) / 256, 256, 0, stream>>>(X3, b3, 200, 1, K3 * HP);

  // ---- up i=0 (j=2): x = X2 + scatter(X3, PERM3); gcn(AN2, u0) relu ----
  hipMemcpyAsync(UA, X2, (size_t)K2 * HP * sizeof(float), hipMemcpyDeviceToDevice, stream);
  k_scatter_add<<<(K3 * HP) / 256, 256, 0, stream>>>(UA, X3, PERM3, K3);
  run_gemm(UA, U0P, XW, K2, HP, HP, HP, HP, HP, stream);
  run_gemm(AN2, XW, UA, K2, HP, K2, K2, HP, HP, stream);
  k_gcn_post<<<(K2 * HP) / 256, 256, 0, stream>>>(UA, u0b, 200, 1, K2 * HP);

  // ---- up i=1 (j=1): x = X1 + scatter(UA, PERM2); gcn(AN1, u1) relu ----
  hipMemcpyAsync(UB, X1, (size_t)K1 * HP * sizeof(float), hipMemcpyDeviceToDevice, stream);
  k_scatter_add<<<(K2 * HP) / 256, 256, 0, stream>>>(UB, UA, PERM2, K2);
  run_gemm(UB, U1P, XW, K1, HP, HP, HP, HP, HP, stream);
  run_gemm(AN1, XW, UB, K1, HP, K1, K1, HP, HP, stream);
  k_gcn_post<<<(K1 * HP) / 256, 256, 0, stream>>>(UB, u1b, 200, 1, K1 * HP);

  // ---- up i=2 (j=0): x = X0 + scatter(UB, PERM1); gcn(AN0, u2); log_softmax ----
  hipMemcpyAsync(UA, X0, (size_t)N0 * HP * sizeof(float), hipMemcpyDeviceToDevice, stream);
  k_scatter_add<<<(K1 * HP) / 256, 256, 0, stream>>>(UA, UB, PERM1, K1);
  run_gemm(UA, U2P, XW, N0, HP, HP, HP, HP, HP, stream);
  run_gemm(AN0, XW, UA, N0, HP, N0, N0, HP, HP, stream);
  k_gcn_post<<<(N0 * HP) / 256, 256, 0, stream>>>(UA, u2b, 2, 0, N0 * HP);
  k_logsoftmax2<<<N0 / 256, 256, 0, stream>>>(UA, out, N0);
}